// AttentionEdgeEncoder_45320494908031
// MI455X (gfx1250) — compile-verified
//
#include <hip/hip_runtime.h>
#include <math.h>

// Problem dims
#define B_ 512
#define N_ 1000
#define D_ 128
#define S_ 8

typedef __attribute__((ext_vector_type(2))) float v2f;
typedef __attribute__((ext_vector_type(8))) float v8f;

// ---------------------------------------------------------------------------
// Kernel 1: gather curr/next embeddings + state projection -> Aq = [B, 3D]
//   Aq[b] = [curr_emb | next_emb | state@w_state^T + b_state]
// ---------------------------------------------------------------------------
__global__ void build_q_input(const float* __restrict__ node_emb,
                              const float* __restrict__ state,
                              const int* __restrict__ curr_id,
                              const int* __restrict__ next_id,
                              const float* __restrict__ w_state,
                              const float* __restrict__ b_state,
                              float* __restrict__ Aq) {
    int b = blockIdx.x;
    int d = threadIdx.x;                       // 128 threads
    const float* ce = node_emb + ((long)b * N_ + curr_id[b]) * D_;
    const float* ne = node_emb + ((long)b * N_ + next_id[b]) * D_;
    float se = b_state[d];
#pragma unroll
    for (int s = 0; s < S_; ++s)
        se += state[b * S_ + s] * w_state[d * S_ + s];
    float* row = Aq + (long)b * (3 * D_);
    row[d]          = ce[d];
    row[D_ + d]     = ne[d];
    row[2 * D_ + d] = se;
}

// ---------------------------------------------------------------------------
// Fully-templated fp32 WMMA GEMM, N fixed to 128.
//   C[m, n] = sum_k A[m*lda + k] * Bsel(k, n)  (+ bias[n]) (+ addm[m*128+n])
//   BMODE 0: Bsel(k,n) = Bm[(k + boff)*128 + n]        (row-major B)
//   BMODE 1: Bsel(k,n) = Bm[(n + boff)*128 + k]        (transposed B)
// Block: 256 threads = 8 waves; wave w owns N-tile [16w, 16w+16).
// Grid: M/16 blocks; block x owns M-tile [16x, 16x+16).
// V_WMMA_F32_16X16X4_F32, full fp32. Inner loop fully unrolled: all branches
// resolved at compile time, base pointers hoisted -> loads use immediate
// offsets and clause ahead of the WMMA chain.
// A frag (16x4 f32): lane l -> m = m0+(l&15), k in {k0+2*(l>>4), +1}
// B frag (4x16 f32): lane l -> n = n0+(l&15), same k pair
// C/D frag: vgpr r -> row = m0 + r + 8*(l>>4), col = n0 + (l&15)
// ---------------------------------------------------------------------------
template <int K, int BMODE, bool HAS_BIAS, bool HAS_ADD>
__global__ void wmma_gemm128(const float* __restrict__ A, int lda,
                             const float* __restrict__ Bm, int boff,
                             const float* __restrict__ bias,
                             const float* __restrict__ addm,
                             float* __restrict__ C) {
    const int m0   = blockIdx.x * 16;
    const int wave = threadIdx.x >> 5;
    const int lane = threadIdx.x & 31;
    const int n0   = wave * 16;
    const int half = lane >> 4;
    const int idx  = lane & 15;
    const int m    = m0 + idx;
    const int n    = n0 + idx;

    const float* pa = A + (size_t)m * lda + 2 * half;
    const float* pb = (BMODE == 0)
        ? Bm + (size_t)(boff + 2 * half) * 128 + n
        : Bm + (size_t)(n + boff) * 128 + 2 * half;

    v8f acc = {};
#pragma unroll
    for (int k0 = 0; k0 < K; k0 += 4) {
        v2f a = *(const v2f*)(pa + k0);
        v2f b;
        if (BMODE == 0) {
            b.x = pb[(size_t)k0 * 128];
            b.y = pb[(size_t)k0 * 128 + 128];
        } else {
            b = *(const v2f*)(pb + k0);
        }
        acc = __builtin_amdgcn_wmma_f32_16x16x4_f32(
            /*neg_a=*/false, a, /*neg_b=*/false, b,
            /*c_mod=*/(short)0, acc, /*reuse_a=*/false, /*reuse_b=*/false);
    }

#pragma unroll
    for (int r = 0; r < 8; ++r) {
        int row = m0 + r + 8 * half;
        float v = acc[r];
        if (HAS_BIAS) v += bias[n];
        if (HAS_ADD)  v += addm[(size_t)row * 128 + n];
        C[(size_t)row * 128 + n] = v;
    }
}

// ---------------------------------------------------------------------------
// Kernel 3: single streaming pass over node_emb with online masked softmax.
//   compat'[b,n] = norm * (node_emb[b,n,:] . kq[b,:])   (constant term cancels)
//   z[b,:] = softmax-weighted sum of node_emb rows.
// One block (8 waves) per batch; one row per wave per iteration; lane holds
// a float4 chunk (coalesced 512B per wave per row). Mask branch is
// wave-uniform. Software-pipelined: next row's load issues before the
// reduce/exp of the current row. Memory-bound: 262 MB @ 23.3 TB/s ~= 11.3 us.
// ---------------------------------------------------------------------------
__global__ void stream_softmax(const float* __restrict__ node_emb,
                               const float* __restrict__ kq,
                               const unsigned char* __restrict__ mask,
                               float* __restrict__ z) {
    const int b    = blockIdx.x;
    const int lane = threadIdx.x & 31;
    const int wave = threadIdx.x >> 5;     // 8 waves
    const float norm = 0.08838834764831845f;  // 1/sqrt(128)

    float4 kqv = *(const float4*)(kq + (size_t)b * D_ + lane * 4);
    const float* base = node_emb + (size_t)b * N_ * D_ + lane * 4;
    const unsigned char* mrow = mask + (size_t)b * N_;

    float m = -INFINITY, s = 0.f;
    float4 zacc = make_float4(0.f, 0.f, 0.f, 0.f);

    float4 e = *(const float4*)(base + (size_t)wave * D_);
    for (int n = wave; n < N_; n += 8) {
        // issue next row's load before consuming the current one
        float4 enext;
        const int nn = n + 8;
        if (nn < N_) {                            // wave-uniform branch
            const float* rp = base + (size_t)nn * D_;
            enext = *(const float4*)rp;
            __builtin_prefetch(rp + 8 * D_, 0, 0);   // global_prefetch_b8
        }

        float p = e.x * kqv.x + e.y * kqv.y + e.z * kqv.z + e.w * kqv.w;
#pragma unroll
        for (int off = 16; off >= 1; off >>= 1)
            p += __shfl_xor(p, off, 32);          // butterfly: all lanes get sum

        if (mrow[n]) {                            // wave-uniform branch
            float x  = norm * p;
            float mn = fmaxf(m, x);
            float sc = __expf(m - mn);
            float pe = __expf(x - mn);
            s = s * sc + pe;
            zacc.x = zacc.x * sc + pe * e.x;
            zacc.y = zacc.y * sc + pe * e.y;
            zacc.z = zacc.z * sc + pe * e.z;
            zacc.w = zacc.w * sc + pe * e.w;
            m = mn;
        }
        e = enext;
    }

    __shared__ float lm[8];
    __shared__ float ls[8];
    __shared__ __align__(16) float lz[8][128];
    ((float4*)(&lz[wave][0]))[lane] = zacc;
    if (lane == 0) { lm[wave] = m; ls[wave] = s; }
    __syncthreads();

    if (threadIdx.x < 128) {
        int d = threadIdx.x;
        float mg = -INFINITY;
#pragma unroll
        for (int w = 0; w < 8; ++w) mg = fmaxf(mg, lm[w]);
        float sg = 0.f, acc = 0.f;
#pragma unroll
        for (int w = 0; w < 8; ++w) {
            float ew = __expf(lm[w] - mg);        // exp(-inf - finite) = 0
            sg  += ls[w] * ew;
            acc += lz[w][d] * ew;
        }
        z[(size_t)b * D_ + d] = acc / sg;
    }
}

// ---------------------------------------------------------------------------
// Launch. Workspace layout (floats): Aq[512*384] q kq vcurr z h  (2 MB total)
// Math (exact restructuring of the reference):
//   q     = Aq @ w_q
//   kq    = q @ Wk2^T            (Wk2 = w_k rows [D,2D))
//   vcurr = curr_emb @ Wv1       (Wv1 = w_v rows [0,D))
//   z     = streaming masked softmax pass
//   h     = vcurr + z @ Wv2      (Wv2 = w_v rows [D,2D))
//   out   = h @ w_out^T + b_out + q
// ---------------------------------------------------------------------------
extern "C" void kernel_launch(void* const* d_in, const int* in_sizes, int n_in,
                              void* d_out, int out_size, void* d_ws, size_t ws_size,
                              hipStream_t stream) {
    const float* node_emb = (const float*)d_in[0];
    const float* state    = (const float*)d_in[1];
    const int*   curr_id  = (const int*)d_in[2];
    const int*   next_id  = (const int*)d_in[3];
    const unsigned char* mask = (const unsigned char*)d_in[4];
    const float* w_q     = (const float*)d_in[5];
    const float* w_k     = (const float*)d_in[6];
    const float* w_v     = (const float*)d_in[7];
    const float* w_state = (const float*)d_in[8];
    const float* b_state = (const float*)d_in[9];
    const float* w_out   = (const float*)d_in[10];
    const float* b_out   = (const float*)d_in[11];
    float* out = (float*)d_out;

    float* ws    = (float*)d_ws;
    float* Aq    = ws;                            // 512*384
    float* q     = Aq    + (size_t)B_ * 3 * D_;   // 512*128
    float* kq    = q     + (size_t)B_ * D_;
    float* vcurr = kq    + (size_t)B_ * D_;
    float* z     = vcurr + (size_t)B_ * D_;
    float* h     = z     + (size_t)B_ * D_;

    build_q_input<<<B_, D_, 0, stream>>>(node_emb, state, curr_id, next_id,
                                         w_state, b_state, Aq);
    // q = Aq[512,384] @ w_q[384,128]
    wmma_gemm128<384, 0, false, false><<<B_ / 16, 256, 0, stream>>>(
        Aq, 3 * D_, w_q, 0, nullptr, nullptr, q);
    // kq = q @ Wk2^T  (B transposed, row offset D)
    wmma_gemm128<128, 1, false, false><<<B_ / 16, 256, 0, stream>>>(
        q, D_, w_k, D_, nullptr, nullptr, kq);
    // vcurr = curr_emb @ Wv1  (curr_emb = first 128 cols of Aq, lda=384)
    wmma_gemm128<128, 0, false, false><<<B_ / 16, 256, 0, stream>>>(
        Aq, 3 * D_, w_v, 0, nullptr, nullptr, vcurr);
    // streaming masked softmax -> z
    stream_softmax<<<B_, 256, 0, stream>>>(node_emb, kq, mask, z);
    // h = vcurr + z @ Wv2
    wmma_gemm128<128, 0, false, true><<<B_ / 16, 256, 0, stream>>>(
        z, D_, w_v, D_, nullptr, vcurr, h);
    // out = h @ w_out^T + b_out + q
    wmma_gemm128<128, 1, true, true><<<B_ / 16, 256, 0, stream>>>(
        h, D_, w_out, 0, b_out, q, out);
}